// BoundaryPathFinder_5093831213734
// MI455X (gfx1250) — compile-verified
//
#include <hip/hip_runtime.h>
#include <math.h>

// ---------------------------------------------------------------------------
// BoundaryPathFinder for MI455X (gfx1250, wave32)
//
// outputs (concatenated in d_out, 4-byte elems):
//   [0, BHW)      grad_map  (float32)
//   [BHW, 2BHW)   seg_init  (int32, bit-stored)
//   [2BHW, 3BHW)  new_seg   (int32, bit-stored)
//
// Roofline: ~240 MB total traffic => ~10us floor @ 23.3 TB/s. Compute is
// negligible; the Sobel's vertical separable factor is mapped onto exact
// FP32 WMMA (V_WMMA_F32_16X16X4_F32, 5 K-chunks) so the matrix pipe carries
// the conv while staying bit-identical to the f32 reference.
// ---------------------------------------------------------------------------

#define Hh 512
#define Ww 512
#define HW (Hh * Ww)

typedef float v2f __attribute__((ext_vector_type(2)));
typedef float v8f __attribute__((ext_vector_type(8)));

// Banded vertical operators U(16x20):  out[i] = sum_a u[a] * S[i + a]
// smooth: u = [1,2,1]   diff: u = [-1,0,1]
__device__ __forceinline__ float u_smooth(int m, int r) {
  int d = r - m;
  return (d == 1) ? 2.0f : ((d == 0 || d == 2) ? 1.0f : 0.0f);
}
__device__ __forceinline__ float u_diff(int m, int r) {
  int d = r - m;
  return (d == 0) ? -1.0f : ((d == 2) ? 1.0f : 0.0f);
}

// ---------------------------------------------------------------------------
// Kernel 1: fused grayscale + Sobel |grad| via WMMA.
// 128 threads = 4 waves; each wave owns one 16x16 output tile.
// Stage A: 18x18 grayscale tile (zero-padded halo, matches conv zero padding)
// Stage B: horizontal 1-D factors into LDS: S_d = diff, S_s = smooth (20x16,
//          rows 18..19 zeroed for the K=20 pad)
// Stage C: vertical factor as 5x V_WMMA_F32_16X16X4_F32 per filter.
// ---------------------------------------------------------------------------
__global__ __launch_bounds__(128) void sobel_wmma_kernel(
    const float* __restrict__ x, float* __restrict__ grad) {
  __shared__ float sh_gray[4][18 * 20];
  __shared__ float sh_sd[4][20 * 16];
  __shared__ float sh_ss[4][20 * 16];

  const int tid  = threadIdx.x;
  const int wv   = tid >> 5;
  const int lane = tid & 31;
  const int b    = blockIdx.z;
  const int i0   = blockIdx.y * 16;
  const int j0   = blockIdx.x * 64 + wv * 16;

  const float* xb = x + (size_t)b * 3 * HW;

  // --- Stage A: grayscale tile with halo (zero pad outside image) ---
  float* gray = sh_gray[wv];
  for (int t = lane; t < 18 * 18; t += 32) {
    int r  = t / 18;
    int cc = t - r * 18;
    int gr = i0 + r - 1;
    int gc = j0 + cc - 1;
    float v = 0.0f;
    if (gr >= 0 && gr < Hh && gc >= 0 && gc < Ww) {
      size_t o = (size_t)gr * Ww + gc;
      v = 0.2989f * xb[o] + 0.587f * xb[o + HW] + 0.114f * xb[o + 2 * (size_t)HW];
    }
    gray[r * 20 + cc] = v;
  }
  __syncthreads();

  // --- Stage B: horizontal separable factors ---
  float* sd = sh_sd[wv];
  float* ss = sh_ss[wv];
  for (int t = lane; t < 20 * 16; t += 32) {
    int r = t >> 4;
    int j = t & 15;
    float vd = 0.0f, vs = 0.0f;
    if (r < 18) {
      float g0 = gray[r * 20 + j];
      float g1 = gray[r * 20 + j + 1];
      float g2 = gray[r * 20 + j + 2];
      vd = g2 - g0;               // [-1,0,1]
      vs = g0 + 2.0f * g1 + g2;   // [ 1,2,1]
    }
    sd[t] = vd;
    ss[t] = vs;
  }
  __syncthreads();

  // --- Stage C: vertical factor on the matrix pipe ---
  // A 16x4 layout: lane m = lane&15; VGPR0 -> K = kb, VGPR1 -> K = kb+1
  // where kb = 4c + (lane<16 ? 0 : 2).  B 4x16: same K split, col = lane&15.
  const int n     = lane & 15;
  const int m     = n;
  const int khalf = (lane < 16) ? 0 : 2;

  v8f accx = {0.f, 0.f, 0.f, 0.f, 0.f, 0.f, 0.f, 0.f};
  v8f accy = {0.f, 0.f, 0.f, 0.f, 0.f, 0.f, 0.f, 0.f};

#pragma unroll
  for (int c = 0; c < 5; ++c) {
    int kb = 4 * c + khalf;
    v2f a_s = {u_smooth(m, kb), u_smooth(m, kb + 1)};
    v2f a_d = {u_diff(m, kb),   u_diff(m, kb + 1)};
    v2f b_d = {sd[kb * 16 + n], sd[(kb + 1) * 16 + n]};
    v2f b_s = {ss[kb * 16 + n], ss[(kb + 1) * 16 + n]};
    // gx = U_smooth * S_diff ; gy = U_diff * S_smooth
    accx = __builtin_amdgcn_wmma_f32_16x16x4_f32(false, a_s, false, b_d,
                                                 (short)0, accx, false, false);
    accy = __builtin_amdgcn_wmma_f32_16x16x4_f32(false, a_d, false, b_s,
                                                 (short)0, accy, false, false);
  }

  // D layout: VGPR g -> row g (lanes 0..15) / g+8 (lanes 16..31), col = n
  float* gout = grad + (size_t)b * HW;
  const int rbase = i0 + ((lane < 16) ? 0 : 8);
#pragma unroll
  for (int g = 0; g < 8; ++g) {
    float gx = accx[g];
    float gy = accy[g];
    gout[(size_t)(rbase + g) * Ww + (j0 + n)] =
        sqrtf(gx * gx + gy * gy + 1e-8f);
  }
}

// ---------------------------------------------------------------------------
// Kernel 2: banded Viterbi DP, one wave per (batch, direction, seam).
// lane n (0..10) owns band position n; neighbor costs via __shfl (wave32).
// Tie-breaking matches jnp.argmin over [left, center, right] (first min).
// Backtrack on lane 0 through the 512x12 LDS predecessor table.
// ---------------------------------------------------------------------------
__global__ __launch_bounds__(32) void dp_kernel(
    const float* __restrict__ grad, int* __restrict__ wsv,
    int* __restrict__ wsh) {
  __shared__ unsigned char path[512 * 12];
  __shared__ float fc[11];

  const int lane = threadIdx.x;
  int blk = blockIdx.x;
  int b   = blk / 14;
  int t   = blk - b * 14;
  int dir = t / 7;        // 0 = vertical seams, 1 = horizontal seams
  int p   = t - dir * 7;
  int base = 64 * (p + 1);
  int cb   = base - 5;    // coord of band position 0 (never clipped: 59..443)

  const float* g = grad + (size_t)b * HW;

  // cost = -grad along the band; lanes 11..31 compute harmless garbage
  float cost = (dir == 0) ? -g[(size_t)0 * Ww + (cb + lane)]
                          : -g[(size_t)(cb + lane) * Ww + 0];

  for (int l = 1; l < 512; ++l) {
    float cy = (dir == 0) ? -g[(size_t)l * Ww + (cb + lane)]
                          : -g[(size_t)(cb + lane) * Ww + l];
    float lf = __shfl(cost, (lane == 0) ? 0 : (lane - 1), 32);
    float rt = __shfl(cost, (lane >= 10) ? 10 : (lane + 1), 32);
    float best = lf;
    int arg = 0;                      // prefer left on ties
    if (cost < best) { best = cost; arg = 1; }
    if (rt   < best) { best = rt;   arg = 2; }
    int nbr = lane + arg - 1;
    nbr = (nbr < 0) ? 0 : ((nbr > 10) ? 10 : nbr);
    if (lane < 11) path[l * 12 + lane] = (unsigned char)nbr;
    cost = best + cy;
  }
  if (lane < 11) fc[lane] = cost;
  __syncthreads();

  if (lane == 0) {
    int idx = 0;
    float bc = fc[0];
    for (int nn = 1; nn < 11; ++nn)
      if (fc[nn] < bc) { bc = fc[nn]; idx = nn; }   // first min, like argmin
    int* outp = ((dir == 0) ? wsv : wsh) + (size_t)b * 512 * 7;
    for (int l = 511; l >= 1; --l) {
      outp[l * 7 + p] = cb + idx;
      idx = path[l * 12 + idx];
    }
    outp[p] = cb + idx;
  }
}

// ---------------------------------------------------------------------------
// Kernel 3: seg_init + new_seg.
// new_seg[b,h,w] = #(vpaths[b,h,:] <= w) + 8 * #(hpaths[b,w,:] <= h)
// (mask scatter + cumsum collapsed to rank counts; seams never collide,
//  spacing 64 >> band 11)
// ---------------------------------------------------------------------------
__global__ __launch_bounds__(512) void label_kernel(
    const int* __restrict__ wsv, const int* __restrict__ wsh,
    int* __restrict__ segi, int* __restrict__ segn) {
  const int h = blockIdx.x;
  const int b = blockIdx.y;
  const int w = threadIdx.x;

  const int* vr = wsv + ((size_t)b * 512 + h) * 7;  // broadcast across block
  const int* hr = wsh + ((size_t)b * 512 + w) * 7;

  int cv = 0, ch = 0;
#pragma unroll
  for (int p = 0; p < 7; ++p) {
    cv += (vr[p] <= w) ? 1 : 0;
    ch += (hr[p] <= h) ? 1 : 0;
  }

  size_t o = ((size_t)b * 512 + h) * 512 + w;
  segi[o] = (h >> 6) * 8 + (w >> 6);
  segn[o] = cv + 8 * ch;
}

// ---------------------------------------------------------------------------
extern "C" void kernel_launch(void* const* d_in, const int* in_sizes, int n_in,
                              void* d_out, int out_size, void* d_ws,
                              size_t ws_size, hipStream_t stream) {
  (void)n_in; (void)out_size; (void)ws_size;

  const float* x = (const float*)d_in[0];
  const int B = in_sizes[0] / (3 * Hh * Ww);   // 32
  const size_t BHW = (size_t)B * HW;

  float* grad = (float*)d_out;                  // region 0: grad_map (f32)
  int* segi = (int*)((float*)d_out + BHW);      // region 1: seg_init (i32)
  int* segn = segi + BHW;                       // region 2: new_seg  (i32)

  int* wsv = (int*)d_ws;                        // [B,512,7] vertical seam cols
  int* wsh = wsv + (size_t)B * 512 * 7;         // [B,512,7] horizontal rows

  dim3 g1(Ww / 64, Hh / 16, B);                 // 4 waves/block, 1 tile/wave
  sobel_wmma_kernel<<<g1, 128, 0, stream>>>(x, grad);

  dp_kernel<<<B * 14, 32, 0, stream>>>(grad, wsv, wsh);

  dim3 g3(Hh, B);
  label_kernel<<<g3, 512, 0, stream>>>(wsv, wsh, segi, segn);
}